// GCNBaseline_17781164606102
// MI455X (gfx1250) — compile-verified
//
#include <hip/hip_runtime.h>
#include <hip/hip_bf16.h>

typedef __attribute__((ext_vector_type(2))) float v2f;
typedef __attribute__((ext_vector_type(8))) float v8f;

// ---------------- utility kernels ----------------

__global__ void zero_f32(float* __restrict__ p, long long n) {
  long long i = (long long)blockIdx.x * blockDim.x + threadIdx.x;
  if (i < n) p[i] = 0.0f;
}

__global__ void deg_init(float* __restrict__ deg, int N) {
  int i = blockIdx.x * blockDim.x + threadIdx.x;
  if (i < N) deg[i] = 1.0f;  // self-loop contribution
}

__global__ void deg_accum(const long long* __restrict__ col, float* __restrict__ deg,
                          long long E) {
  long long e = (long long)blockIdx.x * blockDim.x + threadIdx.x;
  if (e < E) unsafeAtomicAdd(&deg[(int)col[e]], 1.0f);
}

__global__ void deg_rsqrt(float* __restrict__ deg, int N) {
  int i = blockIdx.x * blockDim.x + threadIdx.x;
  if (i < N) deg[i] = rsqrtf(deg[i]);  // deg >= 1 always (self-loops)
}

// ---------------- WMMA GEMM: H[N,64] = X[N,256] @ W[256,64] ----------------
// 128 threads = 4 waves; block computes a 64-row M tile x 64 columns.
// Weights staged in LDS K-pair-interleaved: entry (k/2)*64+n = (W[k][n], W[k+1][n])
// so each B fragment is a single aligned ds_load_b64 (even VGPR pair, no repack movs).

__global__ void __launch_bounds__(128) gemm_l1(const float* __restrict__ X,
                                               const float* __restrict__ W,
                                               float* __restrict__ H, int N) {
  __shared__ v2f Ws[128 * 64];  // 64 KB of the 320 KB WGP LDS
  const int tid = threadIdx.x;
  for (int p = tid; p < 128 * 64; p += 128) {
    const int kp = p >> 6;        // K pair index
    const int n  = p & 63;        // output column
    v2f w;
    w.x = W[(2 * kp) * 64 + n];
    w.y = W[(2 * kp + 1) * 64 + n];
    Ws[p] = w;
  }
  __syncthreads();

  const int wave = tid >> 5;
  const int lane = tid & 31;
  const int lh = lane >> 4;   // half-wave selector (K pair / M+8)
  const int lm = lane & 15;

  int rowA = blockIdx.x * 64 + wave * 16 + lm;
  if (rowA > N - 1) rowA = N - 1;  // clamp reads; stores are guarded
  const float* __restrict__ xr = X + (size_t)rowA * 256;
  const v2f* __restrict__ wb = &Ws[lm];

  v8f a0 = {}, a1 = {}, a2 = {}, a3 = {};
  for (int k0 = 0; k0 < 256; k0 += 4) {
    const int ka = k0 + 2 * lh;          // even
    const int kp = ka >> 1;
    float2 av = *(const float2*)(xr + ka);   // A[m][ka], A[m][ka+1]
    v2f a; a.x = av.x; a.y = av.y;
    const v2f* wk = wb + kp * 64;
    v2f b0 = wk[0];
    v2f b1 = wk[16];
    v2f b2 = wk[32];
    v2f b3 = wk[48];
    a0 = __builtin_amdgcn_wmma_f32_16x16x4_f32(false, a, false, b0, (short)0, a0, false, false);
    a1 = __builtin_amdgcn_wmma_f32_16x16x4_f32(false, a, false, b1, (short)0, a1, false, false);
    a2 = __builtin_amdgcn_wmma_f32_16x16x4_f32(false, a, false, b2, (short)0, a2, false, false);
    a3 = __builtin_amdgcn_wmma_f32_16x16x4_f32(false, a, false, b3, (short)0, a3, false, false);
  }

  const int rowD = blockIdx.x * 64 + wave * 16 + 8 * lh;
#pragma unroll
  for (int r = 0; r < 8; ++r) {
    int row = rowD + r;
    if (row < N) {
      float* hr = H + (size_t)row * 64 + lm;
      hr[0] = a0[r]; hr[16] = a1[r]; hr[32] = a2[r]; hr[48] = a3[r];
    }
  }
}

// ---------------- WMMA GEMM: H[N,32] = X[N,64] @ W[64,32] ----------------

__global__ void __launch_bounds__(128) gemm_l2(const float* __restrict__ X,
                                               const float* __restrict__ W,
                                               float* __restrict__ H, int N) {
  __shared__ v2f Ws[32 * 32];  // 8 KB, K-pair interleaved
  const int tid = threadIdx.x;
  for (int p = tid; p < 32 * 32; p += 128) {
    const int kp = p >> 5;
    const int n  = p & 31;
    v2f w;
    w.x = W[(2 * kp) * 32 + n];
    w.y = W[(2 * kp + 1) * 32 + n];
    Ws[p] = w;
  }
  __syncthreads();

  const int wave = tid >> 5;
  const int lane = tid & 31;
  const int lh = lane >> 4;
  const int lm = lane & 15;

  int rowA = blockIdx.x * 64 + wave * 16 + lm;
  if (rowA > N - 1) rowA = N - 1;
  const float* __restrict__ xr = X + (size_t)rowA * 64;
  const v2f* __restrict__ wb = &Ws[lm];

  v8f a0 = {}, a1 = {};
  for (int k0 = 0; k0 < 64; k0 += 4) {
    const int ka = k0 + 2 * lh;
    const int kp = ka >> 1;
    float2 av = *(const float2*)(xr + ka);
    v2f a; a.x = av.x; a.y = av.y;
    const v2f* wk = wb + kp * 32;
    v2f b0 = wk[0];
    v2f b1 = wk[16];
    a0 = __builtin_amdgcn_wmma_f32_16x16x4_f32(false, a, false, b0, (short)0, a0, false, false);
    a1 = __builtin_amdgcn_wmma_f32_16x16x4_f32(false, a, false, b1, (short)0, a1, false, false);
  }

  const int rowD = blockIdx.x * 64 + wave * 16 + 8 * lh;
#pragma unroll
  for (int r = 0; r < 8; ++r) {
    int row = rowD + r;
    if (row < N) {
      float* hr = H + (size_t)row * 32 + lm;
      hr[0] = a0[r]; hr[16] = a1[r];
    }
  }
}

// ---------------- edge scatter: agg[col] += h[row] * dis[row]*dis[col] ----------------
// F/4 threads per edge, float4 gathers, hardware f32 atomics (L2-resident tables).

template <int F>
__global__ void scatter_edges(const float* __restrict__ Hs,
                              const long long* __restrict__ row,
                              const long long* __restrict__ col,
                              const float* __restrict__ dis,
                              float* __restrict__ agg, long long E) {
  constexpr int TPE = F / 4;
  long long t = (long long)blockIdx.x * blockDim.x + threadIdx.x;
  long long e = t / TPE;
  if (e >= E) return;
  const int c = (int)(t % TPE) * 4;
  const int r  = (int)row[e];
  const int cl = (int)col[e];
  const float nrm = dis[r] * dis[cl];
  const float4 v = *(const float4*)(Hs + (size_t)r * F + c);
  float* dst = agg + (size_t)cl * F + c;
  unsafeAtomicAdd(dst + 0, v.x * nrm);
  unsafeAtomicAdd(dst + 1, v.y * nrm);
  unsafeAtomicAdd(dst + 2, v.z * nrm);
  unsafeAtomicAdd(dst + 3, v.w * nrm);
}

// ---------------- finalize layer1: in-place ELU(agg + h*dis^2 + b) ----------------

__global__ void finalize_l1(float* __restrict__ agg, const float* __restrict__ Hs,
                            const float* __restrict__ dis, const float* __restrict__ b,
                            int N) {
  long long i = (long long)blockIdx.x * blockDim.x + threadIdx.x;
  if (i >= (long long)N * 64) return;
  const int node = (int)(i >> 6);
  const int f = (int)(i & 63);
  const float d = dis[node];
  float v = agg[i] + Hs[i] * d * d + b[f];
  agg[i] = v > 0.0f ? v : expm1f(v);  // ELU alpha=1
}

// ---------------- fused finalize layer2 + readout head ----------------
// warp (wave32) per node: lane f holds feature f; shuffle-reduce the Wc dot.

__global__ void head_fused(const float* __restrict__ agg2, const float* __restrict__ H2,
                           const float* __restrict__ dis, const float* __restrict__ b2,
                           const float* __restrict__ Wc, const float* __restrict__ bc,
                           float* __restrict__ out, int N) {
  const int node = (int)((blockIdx.x * (long long)blockDim.x + threadIdx.x) >> 5);
  const int lane = threadIdx.x & 31;
  if (node >= N) return;
  const float d = dis[node];
  const size_t idx = (size_t)node * 32 + lane;
  float v = agg2[idx] + H2[idx] * d * d + b2[lane];
  v = v > 0.0f ? v : expm1f(v);
  float p = v * Wc[lane];
#pragma unroll
  for (int off = 16; off; off >>= 1) p += __shfl_xor(p, off, 32);
  if (lane == 0) out[node] = p + bc[0];
}

// ---------------- launch ----------------

extern "C" void kernel_launch(void* const* d_in, const int* in_sizes, int n_in,
                              void* d_out, int out_size, void* d_ws, size_t ws_size,
                              hipStream_t stream) {
  const float*     x  = (const float*)d_in[0];
  const long long* ei = (const long long*)d_in[1];  // int64 [2,E]
  const float*     W1 = (const float*)d_in[2];
  const float*     b1 = (const float*)d_in[3];
  const float*     W2 = (const float*)d_in[4];
  const float*     b2 = (const float*)d_in[5];
  const float*     Wc = (const float*)d_in[6];
  const float*     bc = (const float*)d_in[7];

  const int       N = in_sizes[0] / 256;
  const long long E = in_sizes[1] / 2;
  const long long* row = ei;       // source
  const long long* col = ei + E;   // target (aggregation index)

  float* ws   = (float*)d_ws;
  float* dis  = ws;                        // N
  float* h1   = dis  + (size_t)N;          // N*64
  float* agg1 = h1   + (size_t)N * 64;     // N*64 (becomes out1 in place)
  float* h2   = agg1 + (size_t)N * 64;     // N*32
  float* agg2 = h2   + (size_t)N * 32;     // N*32

  const int T = 256;
  auto blk = [](long long n, int t) { return (unsigned)((n + t - 1) / t); };

  // init accumulators + normalization
  zero_f32<<<blk((long long)N * 64, T), T, 0, stream>>>(agg1, (long long)N * 64);
  zero_f32<<<blk((long long)N * 32, T), T, 0, stream>>>(agg2, (long long)N * 32);
  deg_init<<<blk(N, T), T, 0, stream>>>(dis, N);
  deg_accum<<<blk(E, T), T, 0, stream>>>(col, dis, E);
  deg_rsqrt<<<blk(N, T), T, 0, stream>>>(dis, N);

  // layer 1
  gemm_l1<<<blk(N, 64), 128, 0, stream>>>(x, W1, h1, N);
  scatter_edges<64><<<blk(E * 16, T), T, 0, stream>>>(h1, row, col, dis, agg1, E);
  finalize_l1<<<blk((long long)N * 64, T), T, 0, stream>>>(agg1, h1, dis, b1, N);

  // layer 2 + head
  gemm_l2<<<blk(N, 64), 128, 0, stream>>>(agg1, W2, h2, N);
  scatter_edges<32><<<blk(E * 8, T), T, 0, stream>>>(h2, row, col, dis, agg2, E);
  head_fused<<<blk((long long)N * 32, T), T, 0, stream>>>(agg2, h2, dis, b2, Wc, bc,
                                                          (float*)d_out, N);
}